// CombineRadialSpeciesWithCentralSpecies_83021717831723
// MI455X (gfx1250) — compile-verified
//
#include <hip/hip_runtime.h>

typedef __attribute__((ext_vector_type(2))) float v2f;
typedef __attribute__((ext_vector_type(8))) float v8f;

#define NSP 8
#define DK 128
#define NC 256
#define PAD 64                    // bin padding = rowgroup granularity (one wave's M extent)
#define LSTRIDE 132               // LDS row stride (floats): conflict-free A-fragment pattern
#define ROWS_PER_BLOCK 128
#define PERM_BYTES 512
#define DYN_LDS_BYTES (PERM_BYTES + ROWS_PER_BLOCK * LSTRIDE * 4)   // 512 + 67584 = 68096

// d_ws int32 layout:
//  [0..7]   counts per species
//  [8..16]  bin start offsets (off[8] == total padded)
//  [17..24] scatter cursors
//  [25]     total padded rows
//  [32..]   perm[] (padded row index -> original row, -1 = pad sentinel)

__global__ void k_count(const int* __restrict__ sp, int M, int* __restrict__ ws) {
  int i = blockIdx.x * blockDim.x + threadIdx.x;
  if (i < M) atomicAdd(&ws[sp[i]], 1);
}

__global__ void k_prefix(int* __restrict__ ws) {
  if (threadIdx.x != 0 || blockIdx.x != 0) return;
  int off = 0;
  int* perm = ws + 32;
  for (int s = 0; s < NSP; ++s) {
    int c = ws[s];
    int p = (c + PAD - 1) & ~(PAD - 1);   // pad each bin to a whole 64-row group
    ws[8 + s]  = off;                     // bin start
    ws[17 + s] = off;                     // scatter cursor
    for (int i = c; i < p; ++i) perm[off + i] = -1;   // sentinel pad rows
    off += p;
  }
  ws[16] = off;
  ws[25] = off;
}

__global__ void k_scatter(const int* __restrict__ sp, int M, int* __restrict__ ws) {
  int i = blockIdx.x * blockDim.x + threadIdx.x;
  if (i < M) {
    int s = sp[i];
    int pos = atomicAdd(&ws[17 + s], 1);
    ws[32 + pos] = i;
  }
}

// 256 threads = 8 waves = 2 rowgroups (64 rows) x 4 n-tiles (64 cols).
// Block covers 128 rows x 256 cols. Each wave: 4 row-subtiles x 4 col-subtiles,
// B fragments shared across the 4 row-subtiles -> 16 WMMA per 8 B-loads.
__global__ __launch_bounds__(256) void k_gemm(const float* __restrict__ vals,
                                              const float* __restrict__ Wmat,
                                              const int* __restrict__ ws,
                                              float* __restrict__ out) {
  extern __shared__ char dynlds[];
  int*   permlds = (int*)dynlds;
  float* Alds    = (float*)(dynlds + PERM_BYTES);

  const int total   = ws[25];
  const int rowbase = blockIdx.x * ROWS_PER_BLOCK;
  if (rowbase >= total) return;                       // uniform block exit

  const int t = threadIdx.x;

  // ---- Stage A (128 rows x 128 K) into LDS via async-to-LDS DMA ----
  // 2 threads per row, 256 B (16 x b128) each.
  {
    const int row  = t >> 1;
    const int half = t & 1;
    int ridx = (rowbase + row < total) ? ws[32 + rowbase + row] : -1;
    if (half == 0) permlds[row] = ridx;
    const int sr = (ridx >= 0) ? ridx : 0;            // clamp: pad rows load row 0 (never stored)

    unsigned lds_addr = (unsigned)(size_t)(Alds + row * LSTRIDE + half * 64);
    unsigned goff     = (unsigned)((sr * DK + half * 64) * 4);
    unsigned long long gbase = (unsigned long long)(size_t)vals;
#pragma unroll
    for (int q = 0; q < 16; ++q) {
      asm volatile("global_load_async_to_lds_b128 %0, %1, %2 offset:0"
                   :: "v"(lds_addr + q * 16), "v"(goff + q * 16), "s"(gbase)
                   : "memory");
    }
    asm volatile("s_wait_asynccnt 0x0" ::: "memory");
  }
  __syncthreads();

  // ---- Per-wave tile assignment ----
  const int wave = t >> 5;
  const int lane = t & 31;
  const int rg   = wave >> 2;                         // rowgroup 0/1 (64 rows each)
  const int nb   = (wave & 3) * 64;                   // 64-col slice
  const int g    = lane >> 4;
  const int lm   = lane & 15;

  const int tilePos = rowbase + rg * 64;
  if (tilePos >= total) return;                       // after barrier: safe per-wave exit

  // species of this (uniform, 64-row-padded) rowgroup
  int s = 0;
#pragma unroll
  for (int j = 1; j < NSP; ++j)
    if (tilePos >= ws[8 + j]) s = j;

  const float* __restrict__ bp    = Wmat + (long long)s * DK * NC + nb + lm;  // W[s][k][nb+lm]
  const float* __restrict__ arow0 = Alds + (rg * 64 + lm) * LSTRIDE + 2 * g;

  v8f acc[4][4] = {};                                 // [row-subtile][col-subtile]

  // ---- K loop: 16x WMMA f32 16x16x4 per k-step, B shared across row-subtiles ----
#pragma unroll 2
  for (int k = 0; k < DK; k += 4) {
    v2f a[4];
#pragma unroll
    for (int r = 0; r < 4; ++r)
      a[r] = *(const v2f*)(arow0 + r * 16 * LSTRIDE + k);   // ds_load_b64, conflict-free
    const int kb = (k + 2 * g) * NC;
#pragma unroll
    for (int j = 0; j < 4; ++j) {
      v2f b;
      b.x = bp[kb + j * 16];
      b.y = bp[kb + NC + j * 16];
#pragma unroll
      for (int r = 0; r < 4; ++r)
        acc[r][j] = __builtin_amdgcn_wmma_f32_16x16x4_f32(
            false, a[r], false, b, (short)0, acc[r][j], false, false);
    }
  }

  // ---- Store through the permutation (C/D layout: VGPR v -> row v / v+8) ----
#pragma unroll
  for (int r = 0; r < 4; ++r) {
#pragma unroll
    for (int v = 0; v < 8; ++v) {
      const int m = r * 16 + v + g * 8;
      const int rowIdx = permlds[rg * 64 + m];
      if (rowIdx >= 0) {
#pragma unroll
        for (int j = 0; j < 4; ++j)
          out[(long long)rowIdx * NC + nb + j * 16 + lm] = acc[r][j][v];
      }
    }
  }
}

extern "C" void kernel_launch(void* const* d_in, const int* in_sizes, int n_in,
                              void* d_out, int out_size, void* d_ws, size_t ws_size,
                              hipStream_t stream) {
  const float* vals = (const float*)d_in[0];   // [M, 128] fp32
  const int*   sp   = (const int*)d_in[1];     // [M] int32
  const float* Wmat = (const float*)d_in[2];   // [8, 128, 256] fp32
  float* out = (float*)d_out;                  // [M, 256] fp32
  int* wsI = (int*)d_ws;

  const int M = in_sizes[0] / DK;

  hipMemsetAsync(d_ws, 0, 32 * sizeof(int), stream);
  k_count  <<<(M + 255) / 256, 256, 0, stream>>>(sp, M, wsI);
  k_prefix <<<1, 1, 0, stream>>>(wsI);
  k_scatter<<<(M + 255) / 256, 256, 0, stream>>>(sp, M, wsI);

  // allow > 64 KB dynamic LDS (gfx1250: 320 KB per WGP); idempotent, capture-safe
  hipFuncSetAttribute((const void*)k_gemm,
                      hipFuncAttributeMaxDynamicSharedMemorySize, DYN_LDS_BYTES);

  const int maxPadded = M + NSP * (PAD - 1);
  const int blocks = (maxPadded + ROWS_PER_BLOCK - 1) / ROWS_PER_BLOCK;
  k_gemm<<<blocks, 256, DYN_LDS_BYTES, stream>>>(vals, Wmat, wsI, out);
}